// SFTS_22917945492055
// MI455X (gfx1250) — compile-verified
//
#include <hip/hip_runtime.h>

typedef __attribute__((ext_vector_type(2))) float v2f;
typedef __attribute__((ext_vector_type(8))) float v8f;

#define LAYERS 12
#define BB     16
#define HH     12
#define NN     197
#define NPAD   208          // 13 waves * 16 cols
#define KSEL   98           // int(197 * 0.5)
#define NFLAGS (BB * NN)    // 3152 per-(b,n) flag words

// ---------------------------------------------------------------------------
// Zero the (b, n) flag array each call (harness does not re-poison d_ws).
// ---------------------------------------------------------------------------
__global__ void SFTS_init_kernel(unsigned int* __restrict__ flags) {
    int i = blockIdx.x * blockDim.x + threadIdx.x;
    if (i < NFLAGS) flags[i] = 0u;
}

// ---------------------------------------------------------------------------
// One block per (modality, b, h) chain: 13 waves, each owning a 16-column
// tile of the running row-vector. v = row0(attn[11]); v = v @ attn[l],
// l = 10..0, all in f32 via V_WMMA_F32_16X16X4_F32 (branch-free inner loop).
// Then rank-select the top-98 of the 196 scores v[1..196] and OR into the
// global flag array.
// ---------------------------------------------------------------------------
__global__ __launch_bounds__(416) void SFTS_chain_kernel(
    const float* __restrict__ rgb,
    const float* __restrict__ tir,
    unsigned int* __restrict__ flags)
{
    __shared__ float vbuf0[NPAD];
    __shared__ float vbuf1[NPAD];

    const int wg  = blockIdx.x;           // 0..383
    const int mod = wg / (BB * HH);
    const int bh  = wg % (BB * HH);
    const int b   = bh / HH;
    const int h   = bh % HH;

    const float* base = (mod == 0) ? rgb : tir;
    const size_t matElems = (size_t)NN * NN;
    const size_t bhOff    = ((size_t)b * HH + h) * matElems;

    const int tid = threadIdx.x;

    // v <- row 0 of attn[11][b,h], zero-padded to NPAD (padding stays 0
    // forever: every layer's write re-zeros n >= NN).
    {
        const float* m11 = base + ((size_t)11 * BB * HH) * matElems + bhOff;
        for (int k = tid; k < NPAD; k += 416)
            vbuf0[k] = (k < NN) ? m11[k] : 0.0f;
    }
    __syncthreads();

    const int wave  = tid >> 5;        // 0..12 -> column tile
    const int lane  = tid & 31;
    const int lhalf = lane >> 4;       // 0: K rows {ka, ka+1}={k,k+1}; 1: {k+2,k+3}
    const int l15   = lane & 15;
    const int n     = wave * 16 + l15; // output column owned by this lane
    const int ncl   = (n < NN) ? n : (NN - 1);  // clamped load column

    float* vcur  = vbuf0;
    float* vnext = vbuf1;

    for (int layer = 10; layer >= 0; --layer) {
        const float* A = base + ((size_t)layer * BB * HH) * matElems + bhOff;
        // Per-lane base: row 2*lhalf, clamped column. Rows advance by k.
        const float* Ab = A + (size_t)(2 * lhalf) * NN + ncl;
        v8f c = {};
        // Rows 0..195: fully in-bounds -> unconditional, branch-free loads.
        #pragma unroll 4
        for (int k = 0; k < NN - 1; k += 4) {
            v2f av;                       // ds_load_b64 broadcast (8B aligned)
            av.x = vcur[k + 2 * lhalf];
            av.y = vcur[k + 2 * lhalf + 1];
            v2f bv;                       // two coalesced row segments
            bv.x = Ab[(size_t)k * NN];
            bv.y = Ab[(size_t)k * NN + NN];
            c = __builtin_amdgcn_wmma_f32_16x16x4_f32(
                    false, av, false, bv, (short)0, c, false, false);
        }
        // Tail: row 196 (K tile rows 196..199; v[197..199]==0 in LDS).
        {
            v2f av;
            av.x = vcur[(NN - 1) + 2 * lhalf];   // v[196] (half0) / 0 (half1)
            av.y = 0.0f;                          // v[197]/v[199] == 0
            v2f bv;
            bv.x = A[(size_t)(NN - 1) * NN + ncl]; // half1 multiplies this by 0
            bv.y = 0.0f;
            c = __builtin_amdgcn_wmma_f32_16x16x4_f32(
                    false, av, false, bv, (short)0, c, false, false);
        }
        // D layout: VGPR0, lanes 0-15 hold M=0 row, N = l15. Re-zero padding.
        if (lane < 16) vnext[n] = (n < NN) ? c[0] : 0.0f;
        __syncthreads();
        float* t = vcur; vcur = vnext; vnext = t;
    }

    // scores s[j] = vcur[j+1], j = 0..195; select exactly KSEL by stable rank
    // (ties broken toward the lower index, matching jax top_k).
    if (tid < NN - 1) {
        const float sj = vcur[tid + 1];
        int rank = 0;
        for (int m = 0; m < NN - 1; ++m) {
            const float sm = vcur[m + 1];
            rank += ((sm > sj) || ((sm == sj) && (m < tid))) ? 1 : 0;
        }
        if (rank < KSEL) {
            // Reference indexes the length-197 mask with the raw score index
            // (no +1 shift) -- reproduce that exactly.
            atomicOr(&flags[b * NN + tid], 1u);
        }
    }
}

// ---------------------------------------------------------------------------
// (b, n) flag -> one-hot float pair: True -> (0,1), False -> (1,0).
// ---------------------------------------------------------------------------
__global__ void SFTS_finalize_kernel(const unsigned int* __restrict__ flags,
                                     float* __restrict__ out) {
    int i = blockIdx.x * blockDim.x + threadIdx.x;
    if (i < NFLAGS) {
        const bool f = flags[i] != 0u;
        out[2 * i + 0] = f ? 0.0f : 1.0f;
        out[2 * i + 1] = f ? 1.0f : 0.0f;
    }
}

extern "C" void kernel_launch(void* const* d_in, const int* in_sizes, int n_in,
                              void* d_out, int out_size, void* d_ws, size_t ws_size,
                              hipStream_t stream) {
    const float* rgb = (const float*)d_in[0];
    const float* tir = (const float*)d_in[1];
    unsigned int* flags = (unsigned int*)d_ws;   // NFLAGS words
    float* out = (float*)d_out;                  // (16, 197, 2) f32

    SFTS_init_kernel<<<(NFLAGS + 255) / 256, 256, 0, stream>>>(flags);
    SFTS_chain_kernel<<<2 * BB * HH, 416, 0, stream>>>(rgb, tir, flags);
    SFTS_finalize_kernel<<<(NFLAGS + 255) / 256, 256, 0, stream>>>(flags, out);
}